// CrossAttention_74010876445267
// MI455X (gfx1250) — compile-verified
//
#include <hip/hip_runtime.h>
#include <hip/hip_bf16.h>
#include <math.h>

typedef __attribute__((ext_vector_type(2))) float v2f;
typedef __attribute__((ext_vector_type(8))) float v8f;

// ---------------------------------------------------------------------------
// Pointwise (1x1 conv) GEMM:  Y[b,o,n] = sum_c W[o,c] * X[b,c,n]
// One wave per 16(M) x 64(N) tile: 4 f32-WMMA accumulators sharing one A load.
// 32-bit element offsets (slice < 2GB) so the K-loop is add-only addressing.
// A layout: lane%16 = M row, VGPR j holds K = 2*(lane/16)+j.
// B layout: lane%16 = N col, VGPR j holds K = 2*(lane/16)+j.
// C layout: VGPR r holds M = r + 8*(lane/16), N = lane%16.
// ---------------------------------------------------------------------------
__global__ __launch_bounds__(32)
void gemm_pw_kernel(const float* __restrict__ W, const float* __restrict__ X,
                    float* __restrict__ Y, int Cout, int Cin, int Nn) {
  const int lane = threadIdx.x & 31;
  const int half = lane >> 4;
  const int l16  = lane & 15;
  const int n0 = blockIdx.x * 64;
  const int m0 = blockIdx.y * 16;
  const int bb = blockIdx.z;
  const float* __restrict__ Xb = X + (size_t)bb * Cin * Nn;   // batch slice
  float* __restrict__ Yb       = Y + (size_t)bb * Cout * Nn;
  const float* __restrict__ wrow = W + (m0 + l16) * Cin;

  const unsigned nlane = (unsigned)(n0 + l16);
  unsigned xoff = (unsigned)(2 * half) * (unsigned)Nn + nlane; // K=2*half row
  const unsigned stepK = 4u * (unsigned)Nn;

  v8f a0 = {}, a1 = {}, a2 = {}, a3 = {};
  for (int k0 = 0; k0 < Cin; k0 += 4) {
    const v2f a = *(const v2f*)(wrow + k0 + 2 * half);
    v2f b0, b1, b2, b3;
    b0.x = Xb[xoff +  0]; b0.y = Xb[xoff + (unsigned)Nn +  0];
    b1.x = Xb[xoff + 16]; b1.y = Xb[xoff + (unsigned)Nn + 16];
    b2.x = Xb[xoff + 32]; b2.y = Xb[xoff + (unsigned)Nn + 32];
    b3.x = Xb[xoff + 48]; b3.y = Xb[xoff + (unsigned)Nn + 48];
    a0 = __builtin_amdgcn_wmma_f32_16x16x4_f32(false, a, false, b0, (short)0, a0, false, false);
    a1 = __builtin_amdgcn_wmma_f32_16x16x4_f32(false, a, false, b1, (short)0, a1, false, false);
    a2 = __builtin_amdgcn_wmma_f32_16x16x4_f32(false, a, false, b2, (short)0, a2, false, false);
    a3 = __builtin_amdgcn_wmma_f32_16x16x4_f32(false, a, false, b3, (short)0, a3, false, false);
    xoff += stepK;
  }
#pragma unroll
  for (int r = 0; r < 8; ++r) {
    const unsigned yoff = (unsigned)(m0 + r + 8 * half) * (unsigned)Nn + nlane;
    Yb[yoff +  0] = a0[r];
    Yb[yoff + 16] = a1[r];
    Yb[yoff + 32] = a2[r];
    Yb[yoff + 48] = a3[r];
  }
}

// ---------------------------------------------------------------------------
// Depthwise 3x3 conv, SAME padding (cross-correlation, matching XLA conv).
// X,Y: [B, Cchan, 128, 128], Wd: [Cchan, 1, 3, 3]
// ---------------------------------------------------------------------------
__global__ __launch_bounds__(256)
void dwconv3_kernel(const float* __restrict__ X, const float* __restrict__ Wd,
                    float* __restrict__ Y, int Cchan, int total) {
  const int gid = blockIdx.x * 256 + threadIdx.x;
  if (gid >= total) return;
  const int x = gid & 127;
  const int y = (gid >> 7) & 127;
  const int cn = gid >> 14;            // bb*Cchan + c
  const int c = cn % Cchan;
  const float* w = Wd + c * 9;
  const float* in = X + (size_t)cn * 16384;
  float s = 0.f;
#pragma unroll
  for (int dy = -1; dy <= 1; ++dy) {
    const int yy = y + dy;
    if (yy < 0 || yy > 127) continue;
#pragma unroll
    for (int dx = -1; dx <= 1; ++dx) {
      const int xx = x + dx;
      if (xx < 0 || xx > 127) continue;
      s += in[yy * 128 + xx] * w[(dy + 1) * 3 + (dx + 1)];
    }
  }
  Y[gid] = s;
}

// ---------------------------------------------------------------------------
// Per-row L2 norm over N: Inv[b*192+c] = 1 / max(sqrt(sum x^2), 1e-12)
// Rows live inside a [b, Ctot, N] tensor at channel c (c < 192).
// ---------------------------------------------------------------------------
__global__ __launch_bounds__(256)
void rownorm_kernel(const float* __restrict__ X, float* __restrict__ Inv,
                    int Nn, int Ctot) {
  const int row = blockIdx.x;          // 0 .. B*192-1
  const int bb = row / 192, c = row % 192;
  const float* p = X + ((size_t)bb * Ctot + c) * Nn;
  float s = 0.f;
  for (int i = threadIdx.x; i < Nn; i += 256) {
    const float v = p[i];
    s += v * v;
  }
  __shared__ float red[256];
  red[threadIdx.x] = s;
  __syncthreads();
  for (int off = 128; off > 0; off >>= 1) {
    if ((int)threadIdx.x < off) red[threadIdx.x] += red[threadIdx.x + off];
    __syncthreads();
  }
  if (threadIdx.x == 0) {
    const float n = sqrtf(red[0]);
    Inv[row] = 1.0f / fmaxf(n, 1e-12f);
  }
}

__global__ __launch_bounds__(256)
void zero_kernel(float* __restrict__ p, int n) {
  const int i = blockIdx.x * 256 + threadIdx.x;
  if (i < n) p[i] = 0.f;
}

// ---------------------------------------------------------------------------
// Attention logits: acc[bh, i, j] += (q_i . k_j) * invq[i] * invk[j]
// Raw dot products accumulated with WMMA (split-K + atomic adds); the two
// normalization scales factorize, so they're applied once in the epilogue
// (padded rows/cols get scale 0). Row indices clamped to stay in-bounds.
// grid: (ksplit, 4 tiles, b*h)
// ---------------------------------------------------------------------------
__global__ __launch_bounds__(32)
void attn_qk_kernel(const float* __restrict__ Q, const float* __restrict__ K,
                    const float* __restrict__ invq, const float* __restrict__ invk,
                    float* __restrict__ Acc, int Nn, int chunk) {
  const int lane = threadIdx.x & 31;
  const int half = lane >> 4;
  const int l16  = lane & 15;
  const int mt = blockIdx.y & 1, nt = blockIdx.y >> 1;
  const int bh = blockIdx.z;
  const int bb = bh >> 3, h = bh & 7;
  const int i = mt * 16 + l16;                   // q channel within head
  const int j = nt * 16 + l16;                   // k channel within head
  const int ic = i < 24 ? i : 23;                // clamp for in-bounds loads
  const int jc = j < 24 ? j : 23;
  const int kstart = blockIdx.x * chunk;

  unsigned qoff = (unsigned)(bb * 192 + h * 24 + ic) * (unsigned)Nn
                  + (unsigned)(kstart + 2 * half);
  unsigned koff = (unsigned)(bb * 384 + h * 24 + jc) * (unsigned)Nn
                  + (unsigned)(kstart + 2 * half);  // k = first 192 ch of kv_dw

  v8f acc = {};
  for (int k0 = 0; k0 < chunk; k0 += 4) {
    const v2f a = *(const v2f*)(Q + qoff);
    const v2f b = *(const v2f*)(K + koff);
    acc = __builtin_amdgcn_wmma_f32_16x16x4_f32(false, a, false, b,
                                                (short)0, acc, false, false);
    qoff += 4; koff += 4;
  }

  const float ksc = (j < 24) ? invk[bb * 192 + h * 24 + j] : 0.f;
  float* out = Acc + (size_t)bh * 1024;
#pragma unroll
  for (int r = 0; r < 8; ++r) {
    const int m = mt * 16 + r + 8 * half;
    const float qsc = (m < 24) ? invq[bb * 192 + h * 24 + m] : 0.f;
    atomicAdd(out + m * 32 + nt * 16 + l16, acc[r] * qsc * ksc);
  }
}

// ---------------------------------------------------------------------------
// Softmax over j (24) with temperature, one wave per (b,h,i) row.
// ---------------------------------------------------------------------------
__global__ __launch_bounds__(32)
void softmax_kernel(float* __restrict__ Acc, const float* __restrict__ temp) {
  const int row = blockIdx.x;          // 0 .. 64*24-1
  const int bh = row / 24, i = row % 24;
  const int h = bh & 7;
  float* p = Acc + (size_t)bh * 1024 + i * 32;
  const int j = threadIdx.x;
  const float t = temp[h];
  const float v = (j < 24) ? p[j] * t : -3.4e38f;
  float m = v;
  for (int off = 16; off > 0; off >>= 1) m = fmaxf(m, __shfl_xor(m, off, 32));
  const float e = (j < 24) ? expf(v - m) : 0.f;
  float s = e;
  for (int off = 16; off > 0; off >>= 1) s += __shfl_xor(s, off, 32);
  if (j < 24) p[j] = e / s;
}

// ---------------------------------------------------------------------------
// out[b, h*24+i, n] = sum_j attn[bh,i,j] * v[b, 192 + h*24+j, n]  (K=24)
// One wave per 16(M) x 64(N) tile, 4 accumulators. Padded attn rows (24..31)
// are exact zeros in the 32x32 accumulator, so A loads need no guard; stores
// are guarded to i<24 so padded rows never clobber the next head.
// grid: (N/64, 2 m-tiles, b*h)
// ---------------------------------------------------------------------------
__global__ __launch_bounds__(32)
void attn_v_kernel(const float* __restrict__ Acc, const float* __restrict__ V,
                   float* __restrict__ Y, int Nn) {
  const int lane = threadIdx.x & 31;
  const int half = lane >> 4;
  const int l16  = lane & 15;
  const int n0 = blockIdx.x * 64;
  const int mt = blockIdx.y;
  const int bh = blockIdx.z;
  const int bb = bh >> 3, h = bh & 7;
  const float* __restrict__ A  = Acc + (size_t)bh * 1024;
  const float* __restrict__ Vb = V + (size_t)bb * 384 * Nn;   // kv_dw batch slice
  float* __restrict__ Yb       = Y + (size_t)bb * 192 * Nn;
  const int i = mt * 16 + l16;

  const unsigned nlane = (unsigned)(n0 + l16);
  unsigned voff = (unsigned)(192 + h * 24 + 2 * half) * (unsigned)Nn + nlane;
  const unsigned stepK = 4u * (unsigned)Nn;

  v8f a0 = {}, a1 = {}, a2 = {}, a3 = {};
  for (int k0 = 0; k0 < 24; k0 += 4) {
    const v2f a = *(const v2f*)(A + i * 32 + k0 + 2 * half);  // rows>=24 are 0
    v2f b0, b1, b2, b3;
    b0.x = Vb[voff +  0]; b0.y = Vb[voff + (unsigned)Nn +  0];
    b1.x = Vb[voff + 16]; b1.y = Vb[voff + (unsigned)Nn + 16];
    b2.x = Vb[voff + 32]; b2.y = Vb[voff + (unsigned)Nn + 32];
    b3.x = Vb[voff + 48]; b3.y = Vb[voff + (unsigned)Nn + 48];
    a0 = __builtin_amdgcn_wmma_f32_16x16x4_f32(false, a, false, b0, (short)0, a0, false, false);
    a1 = __builtin_amdgcn_wmma_f32_16x16x4_f32(false, a, false, b1, (short)0, a1, false, false);
    a2 = __builtin_amdgcn_wmma_f32_16x16x4_f32(false, a, false, b2, (short)0, a2, false, false);
    a3 = __builtin_amdgcn_wmma_f32_16x16x4_f32(false, a, false, b3, (short)0, a3, false, false);
    voff += stepK;
  }
#pragma unroll
  for (int r = 0; r < 8; ++r) {
    const int m = mt * 16 + r + 8 * half;
    if (m < 24) {
      const unsigned yoff = (unsigned)(h * 24 + m) * (unsigned)Nn + nlane;
      Yb[yoff +  0] = a0[r];
      Yb[yoff + 16] = a1[r];
      Yb[yoff + 32] = a2[r];
      Yb[yoff + 48] = a3[r];
    }
  }
}

extern "C" void kernel_launch(void* const* d_in, const int* in_sizes, int n_in,
                              void* d_out, int out_size, void* d_ws, size_t ws_size,
                              hipStream_t stream) {
  (void)in_sizes; (void)n_in; (void)out_size; (void)ws_size;
  const float* x_main      = (const float*)d_in[0];
  const float* x_aux       = (const float*)d_in[1];
  const float* w_qkv_main  = (const float*)d_in[2];
  const float* w_dw_main   = (const float*)d_in[3];
  const float* w_qkv_aux   = (const float*)d_in[4];
  const float* w_dw_aux    = (const float*)d_in[5];
  const float* w_proj      = (const float*)d_in[6];
  const float* temperature = (const float*)d_in[7];
  float* out = (float*)d_out;

  // workspace layout (floats)
  float* ws   = (float*)d_ws;
  float* kv   = ws;                         //  8*384*16384 = 50331648
  float* kvdw = ws + 50331648;              //  50331648
  float* q    = ws + 100663296;             //  8*192*16384 = 25165824  (reused as attn_out)
  float* qdw  = ws + 125829120;             //  25165824
  float* acc  = ws + 150994944;             //  64*32*32 = 65536 (padded logits)
  float* invq = acc + 65536;                //  8*192
  float* invk = invq + 1536;                //  8*192

  const int Nn = 16384;

  // 0. zero the attention accumulator (split-K atomics land here)
  zero_kernel<<<(65536 + 255) / 256, 256, 0, stream>>>(acc, 65536);

  // 1. kv = w_qkv_main (384x192) @ main      -> kv [8,384,N]
  gemm_pw_kernel<<<dim3(256, 24, 8), 32, 0, stream>>>(w_qkv_main, x_main, kv, 384, 192, Nn);

  // 2. q  = w_qkv_aux (192x192) @ aux        -> q  [8,192,N]
  gemm_pw_kernel<<<dim3(256, 12, 8), 32, 0, stream>>>(w_qkv_aux, x_aux, q, 192, 192, Nn);

  // 3. depthwise 3x3 on kv and q
  dwconv3_kernel<<<50331648 / 256, 256, 0, stream>>>(kv, w_dw_main, kvdw, 384, 50331648);
  dwconv3_kernel<<<25165824 / 256, 256, 0, stream>>>(q,  w_dw_aux,  qdw, 192, 25165824);

  // 4. L2-norm scales over spatial axis for q and k
  rownorm_kernel<<<1536, 256, 0, stream>>>(qdw,  invq, Nn, 192);
  rownorm_kernel<<<1536, 256, 0, stream>>>(kvdw, invk, Nn, 384);

  // 5. logits = qhat @ khat^T, split-K (64 chunks of 256) + atomic accumulate
  attn_qk_kernel<<<dim3(64, 4, 64), 32, 0, stream>>>(qdw, kvdw, invq, invk, acc, Nn, 256);

  // 6. temperature + softmax over the 24-wide rows
  softmax_kernel<<<1536, 32, 0, stream>>>(acc, temperature);

  // 7. attn @ v  -> reuse q buffer as attention output [8,192,N]
  attn_v_kernel<<<dim3(256, 2, 64), 32, 0, stream>>>(acc, kvdw, q, Nn);

  // 8. final projection: out = w_proj (192x192) @ attn_out
  gemm_pw_kernel<<<dim3(256, 12, 8), 32, 0, stream>>>(w_proj, q, out, 192, 192, Nn);
}